// APDA_26061861552908
// MI455X (gfx1250) — compile-verified
//
#include <hip/hip_runtime.h>
#include <math.h>

#define N_USERS 100000
#define N_ITEMS 50000
#define N_NODES 150000
#define N_EDGES 2000000
#define DIM     64
#define RESID   0.2f
#define INV_EXPC 0.2f  /* 1/EXP_COFF */

typedef __attribute__((ext_vector_type(2))) float v2f;
typedef __attribute__((ext_vector_type(8))) float v8f;

__device__ __forceinline__ float wave_sum32(float v) {
#pragma unroll
  for (int off = 16; off >= 1; off >>= 1) v += __shfl_xor(v, off, 32);
  return v;
}

__device__ __forceinline__ const float* orig_row(const float* UE, const float* IE, int node) {
  return (node < N_USERS) ? (UE + (size_t)node * DIM)
                          : (IE + (size_t)(node - N_USERS) * DIM);
}

// acc(out) = orig ; B0 = orig ; invn = 1/||orig||
__global__ void __launch_bounds__(256) k_init(const float* __restrict__ UE,
                                              const float* __restrict__ IE,
                                              float* __restrict__ B0,
                                              float* __restrict__ out,
                                              float* __restrict__ invn) {
  int wid  = (blockIdx.x * 256 + threadIdx.x) >> 5;   // node id
  int lane = threadIdx.x & 31;
  const float* op = orig_row(UE, IE, wid);
  float2 o = *(const float2*)(op + lane * 2);
  float s = wave_sum32(o.x * o.x + o.y * o.y);
  float iv = 1.0f / fmaxf(sqrtf(s), 1e-12f);
  size_t base = (size_t)wid * DIM + lane * 2;
  *(float2*)(B0 + base)  = o;
  *(float2*)(out + base) = o;
  if (lane == 0) invn[wid] = iv;
}

// cur = l2norm(cur + 0.2*initial) in place; zero the neighbor buffer
__global__ void __launch_bounds__(256) k_prenorm(float* __restrict__ cur,
                                                 float* __restrict__ oth,
                                                 const float* __restrict__ UE,
                                                 const float* __restrict__ IE,
                                                 const float* __restrict__ invn) {
  int wid  = (blockIdx.x * 256 + threadIdx.x) >> 5;
  int lane = threadIdx.x & 31;
  const float* op = orig_row(UE, IE, wid);
  float2 o = *(const float2*)(op + lane * 2);
  float iv = invn[wid];
  size_t base = (size_t)wid * DIM + lane * 2;
  float2 r = *(float2*)(cur + base);
  r.x += RESID * o.x * iv;
  r.y += RESID * o.y * iv;
  float s = wave_sum32(r.x * r.x + r.y * r.y);
  float n = fmaxf(sqrtf(s), 1e-12f);
  r.x /= n; r.y /= n;
  *(float2*)(cur + base) = r;
  float2 z; z.x = 0.0f; z.y = 0.0f;
  *(float2*)(oth + base) = z;
}

// Edge pass: 16 edges per wave.
//  - dots for all 16 edges via 16 accumulating V_WMMA_F32_16X16X4_F32
//  - diagonal extracted to the edge-owning lane (8 bpermute + 8 select)
//  - exp/softplus computed once per lane (16 edges in parallel)
//  - scatter: wave-uniform indices via scalar loads, mat via 1 bpermute/edge,
//    64 f32 atomics per edge (2 per lane)
__global__ void __launch_bounds__(256) k_edges(const float* __restrict__ emb,
                                               float* __restrict__ nbr,
                                               const int* __restrict__ src,
                                               const int* __restrict__ dst,
                                               const float* __restrict__ val) {
  int wid   = (blockIdx.x * 256 + threadIdx.x) >> 5;  // wave id
  int lane  = threadIdx.x & 31;
  int e     = lane & 15;                              // edge slot within wave
  int kslot = (lane >> 4) * 2;                        // K offset within 4-chunk
  int E0    = wid * 16;
  int E0u   = __builtin_amdgcn_readfirstlane(E0);     // provably wave-uniform

  int   s = src[E0 + e];
  int   d = dst[E0 + e];
  float w = val[E0 + e];

  const float* srow = emb + (size_t)s * DIM;
  const float* drow = emb + (size_t)d * DIM;

  v8f c = {};  // C[i][j] accumulates dot(src_i, dst_j); we consume the diagonal
#pragma unroll
  for (int kb = 0; kb < DIM; kb += 4) {
    v2f a = *(const v2f*)(srow + kb + kslot);  // A 16x4 frag: lane=M, VGPRs=K pair
    v2f b = *(const v2f*)(drow + kb + kslot);  // B 4x16 frag: lane=N, VGPRs=K pair
    c = __builtin_amdgcn_wmma_f32_16x16x4_f32(false, a, false, b, (short)0, c,
                                              false, false);
  }

  // Bring diag C[e][e] to lane e (and its mirror lane e+16).
  // C[e][e] lives in accumulator component e&7, on lane e (e<8) or e+16 (e>=8).
  int diagSrc = (e & 8) ? (e + 16) : e;
  float dot = 0.0f;
#pragma unroll
  for (int r = 0; r < 8; ++r) {
    float g = __shfl(c[r], diagSrc, 32);
    if ((e & 7) == r) dot = g;
  }

  // Per-lane edge weight: one exp + one log1p per lane (16 edges in parallel).
  float cp  = dot * (1.0f / DIM);
  float t   = (2.0f - 2.0f * cp) * INV_EXPC;
  float et  = expf(t);
  float mat = 0.5f * et * log1pf(et) * w;     // 0.5*exp(t)*softplus(t)*value

#pragma unroll
  for (int e2 = 0; e2 < 16; ++e2) {
    int   sb = src[E0u + e2];                 // wave-uniform -> s_load path
    int   db = dst[E0u + e2];
    float mb = __shfl(mat, e2, 32);           // edge e2's weight from lane e2

    float2 dv = *(const float2*)(emb + (size_t)db * DIM + lane * 2);
    float* o  = nbr + (size_t)sb * DIM + lane * 2;
    atomicAdd(o + 0, dv.x * mb);
    atomicAdd(o + 1, dv.y * mb);
  }
}

// oth = neighbor + 0.2*(cur - initial) ; out += oth
__global__ void __launch_bounds__(256) k_post(float* __restrict__ oth,
                                              const float* __restrict__ cur,
                                              const float* __restrict__ UE,
                                              const float* __restrict__ IE,
                                              const float* __restrict__ invn,
                                              float* __restrict__ out) {
  int wid  = (blockIdx.x * 256 + threadIdx.x) >> 5;
  int lane = threadIdx.x & 31;
  const float* op = orig_row(UE, IE, wid);
  float2 o = *(const float2*)(op + lane * 2);
  float iv = invn[wid];
  size_t base = (size_t)wid * DIM + lane * 2;
  float2 cc = *(const float2*)(cur + base);
  float2 v  = *(float2*)(oth + base);
  v.x += RESID * (cc.x - o.x * iv);
  v.y += RESID * (cc.y - o.y * iv);
  *(float2*)(oth + base) = v;
  float2 a = *(float2*)(out + base);
  a.x += v.x; a.y += v.y;
  *(float2*)(out + base) = a;
}

__global__ void __launch_bounds__(256) k_scale(float* __restrict__ out) {
  size_t i = ((size_t)blockIdx.x * 256 + threadIdx.x) * 4;
  float4 v = *(float4*)(out + i);
  v.x *= 0.25f; v.y *= 0.25f; v.z *= 0.25f; v.w *= 0.25f;
  *(float4*)(out + i) = v;
}

extern "C" void kernel_launch(void* const* d_in, const int* in_sizes, int n_in,
                              void* d_out, int out_size, void* d_ws, size_t ws_size,
                              hipStream_t stream) {
  const float* UE = (const float*)d_in[0];
  const float* IE = (const float*)d_in[1];
  const int*   EI = (const int*)d_in[2];
  const float* EV = (const float*)d_in[3];
  const int* src = EI;
  const int* dst = EI + N_EDGES;
  float* out = (float*)d_out;

  char* ws = (char*)d_ws;
  const size_t EMB_BYTES = (size_t)N_NODES * DIM * sizeof(float);  // 38.4 MB
  float* B0   = (float*)(ws);
  float* B1   = (float*)(ws + EMB_BYTES);
  float* invn = (float*)(ws + 2 * EMB_BYTES);

  dim3 tb(256);
  dim3 nodeGrid(N_NODES / 8);            // 8 waves/block, 1 node/wave -> 18750
  dim3 edgeGrid(N_EDGES / 16 / 8);       // 16 edges/wave, 8 waves/block -> 15625
  dim3 scaleGrid((N_NODES * DIM) / 4 / 256);  // 9375

  k_init<<<nodeGrid, tb, 0, stream>>>(UE, IE, B0, out, invn);

  float* cur = B0;
  float* oth = B1;
  for (int layer = 0; layer < 3; ++layer) {
    k_prenorm<<<nodeGrid, tb, 0, stream>>>(cur, oth, UE, IE, invn);
    k_edges<<<edgeGrid, tb, 0, stream>>>(cur, oth, src, dst, EV);
    k_post<<<nodeGrid, tb, 0, stream>>>(oth, cur, UE, IE, invn, out);
    float* t = cur; cur = oth; oth = t;
  }

  k_scale<<<scaleGrid, tb, 0, stream>>>(out);
}